// SmoothRankAP_30374008717321
// MI455X (gfx1250) — compile-verified
//
#include <hip/hip_runtime.h>

#define BQ 512
#define NWAVES 8

typedef float v2f __attribute__((ext_vector_type(2)));
typedef float v8f __attribute__((ext_vector_type(8)));

// log2(e) / TAU, TAU = 0.01
#define SIG_K 144.26950408889634f

// sigmoid((sj - si)/TAU) where inputs are pre-scaled by SIG_K:
//   1 / (1 + 2^(siK - sjK))
static __device__ __forceinline__ float sig_scaled(float siK, float sjK) {
  float e = __builtin_amdgcn_exp2f(siK - sjK);
  return __builtin_amdgcn_rcpf(1.0f + e);
}

static __device__ __forceinline__ float read_lane_f32(float v, int src) {
  return __builtin_bit_cast(float,
      __builtin_amdgcn_readlane(__builtin_bit_cast(int, v), src));
}

// One block per query row q. Computes ap[q] into ap_out[q].
__global__ __launch_bounds__(NWAVES * 32) void smoothap_ap_kernel(
    const float* __restrict__ scores,
    const float* __restrict__ target,
    float* __restrict__ ap_out) {
  __shared__ float sK_row[BQ];         // scores[q, :] * SIG_K
  __shared__ float w_row[BQ];          // target[q, :] - (j == q)   (pos_mask row)
  __shared__ int   poslist[BQ];        // indices i with target[q,i]==1, padded to 16
  __shared__ int   s_npos;
  __shared__ float s_all[NWAVES][16];
  __shared__ float s_pos[NWAVES][16];
  __shared__ float s_red[16];

  const int q    = blockIdx.x;
  const int tid  = threadIdx.x;
  const int lane = tid & 31;
  // wave id is wave-uniform: force it into an SGPR so the j-loop is scalar
  const int wave = __builtin_amdgcn_readfirstlane(tid >> 5);

  // Stage the q-th rows (pre-scaled scores, pos_mask) into LDS.
  for (int j = tid; j < BQ; j += NWAVES * 32) {
    float s = scores[q * BQ + j];
    float t = target[q * BQ + j];
    sK_row[j] = s * SIG_K;
    w_row[j]  = t - ((j == q) ? 1.0f : 0.0f);
  }
  __syncthreads();

  // Deterministic order-preserving compaction of positive columns (wave 0).
  if (wave == 0) {
    int count = 0;
    for (int base = 0; base < BQ; base += 32) {
      int j = base + lane;
      float t = (j == q) ? 1.0f : w_row[j];
      bool p = t > 0.5f;
      unsigned mask = __builtin_amdgcn_ballot_w32(p);
      int prefix = __builtin_popcount(mask & ((1u << lane) - 1u));
      if (p) poslist[count + prefix] = j;
      count += __builtin_popcount(mask);
    }
    if (lane == 0) {
      s_npos = count;
      int npad = (count + 15) & ~15;
      for (int j = count; j < npad; ++j) poslist[j] = poslist[0];  // dummy rows, discarded
    }
  }
  __syncthreads();

  const int npos    = s_npos;              // >= 1 (diagonal is always positive)
  const int ngroups = (npos + 15) >> 4;

  // WMMA f32 16x16x4 fragment geometry (wave32):
  //   A (16x4): lane L -> row M = L&15, K = (L<16 ? {0,1} : {2,3})
  //   B (4x16): lane L -> col N = L&15, same K split
  //   C/D     : column n lives in lanes {n, n+16}, rows split across the 8 VGPRs
  const int m     = lane & 15;
  const int khalf = (lane < 16) ? 0 : 2;
  // Loop-invariant selection masks for the B fragment (branch-free build).
  const float is0 = (m == 0) ? 1.0f : 0.0f;   // column 0: ones  -> rank-all sum
  const float is1 = (m == 1) ? 1.0f : 0.0f;   // column 1: w[j]  -> pos-rank sum

  const v2f* __restrict__ sK2 = (const v2f*)sK_row;  // paired (b64) LDS reads
  const v2f* __restrict__ w2  = (const v2f*)w_row;

  float my_ap = 0.0f;  // meaningful on tid < 16

  for (int ig = 0; ig < ngroups; ++ig) {
    const int   i_idx = poslist[ig * 16 + m];
    const float siK   = sK_row[i_idx];

    v8f acc = {};  // D[:,0] = sum_j sg ; D[:,1] = sum_j sg*w ; others 0

    // Each wave covers a strided slice of the j range, 4 K-chunks (=16 j) per step.
    for (int jb = wave * 16; jb < BQ; jb += NWAVES * 16) {
#pragma unroll
      for (int c = 0; c < 4; ++c) {
        const int p = (jb + c * 4 + khalf) >> 1;   // float2 index (always aligned)
        const v2f sj = sK2[p];
        const v2f wj = w2[p];
        v2f a, b;
        a.x = sig_scaled(siK, sj.x);
        a.y = sig_scaled(siK, sj.y);
        b.x = __builtin_fmaf(is1, wj.x, is0);
        b.y = __builtin_fmaf(is1, wj.y, is0);
        acc = __builtin_amdgcn_wmma_f32_16x16x4_f32(
            /*neg_a=*/false, a, /*neg_b=*/false, b,
            /*c_mod=*/(short)0, acc, /*reuse_a=*/false, /*reuse_b=*/false);
      }
    }

    // Extract D columns 0 (all-rank sum) and 1 (pos-rank sum) for the 16 rows.
#pragma unroll
    for (int mm = 0; mm < 16; ++mm) {
      const int src = (mm < 8) ? 0 : 16;
      float av = read_lane_f32(acc[mm & 7], src);
      float pv = read_lane_f32(acc[mm & 7], src + 1);
      if (lane == 0) { s_all[wave][mm] = av; s_pos[wave][mm] = pv; }
    }
    __syncthreads();

    if (tid < 16) {
      // all_rk = (rowsum - sg(i,i)=0.5) + 1 ; pos_rk = pos_sum + target[q,i](=1)
      float av = 0.5f;
      float pv = 1.0f;
#pragma unroll
      for (int w2i = 0; w2i < NWAVES; ++w2i) { av += s_all[w2i][tid]; pv += s_pos[w2i][tid]; }
      bool valid = (ig * 16 + tid) < npos;
      my_ap += valid ? (pv / av) : 0.0f;
    }
    __syncthreads();  // protect s_all/s_pos reuse next group
  }

  if (tid < 16) s_red[tid] = my_ap;
  __syncthreads();
  if (tid == 0) {
    float total = 0.0f;
#pragma unroll
    for (int k = 0; k < 16; ++k) total += s_red[k];
    ap_out[q] = total / (float)npos;
  }
}

// Final reduction: out = 1 - mean(ap)
__global__ __launch_bounds__(256) void smoothap_finalize(
    const float* __restrict__ ap, float* __restrict__ out) {
  __shared__ float red[256];
  int tid = threadIdx.x;
  red[tid] = ap[tid] + ap[tid + 256];
  __syncthreads();
  for (int s = 128; s > 0; s >>= 1) {
    if (tid < s) red[tid] += red[tid + s];
    __syncthreads();
  }
  if (tid == 0) out[0] = 1.0f - red[0] * (1.0f / (float)BQ);
}

extern "C" void kernel_launch(void* const* d_in, const int* in_sizes, int n_in,
                              void* d_out, int out_size, void* d_ws, size_t ws_size,
                              hipStream_t stream) {
  (void)in_sizes; (void)n_in; (void)out_size; (void)ws_size;
  const float* scores = (const float*)d_in[0];
  const float* target = (const float*)d_in[1];
  float* ap_ws = (float*)d_ws;  // 512 floats of scratch

  smoothap_ap_kernel<<<BQ, NWAVES * 32, 0, stream>>>(scores, target, ap_ws);
  smoothap_finalize<<<1, 256, 0, stream>>>(ap_ws, (float*)d_out);
}